// NoisyTopKRouter_33921651704703
// MI455X (gfx1250) — compile-verified
//
#include <hip/hip_runtime.h>
#include <hip/hip_bf16.h>

typedef float v2f __attribute__((ext_vector_type(2)));
typedef float v4f __attribute__((ext_vector_type(4)));
typedef float v8f __attribute__((ext_vector_type(8)));

#define H_DIM 4096
#define E_DIM 8
#define KHALF 2048
#define CHUNK 512
#define NTOK  16384
#define NOISE_EPS 0.01f
#define LB_COEFF  0.01f

// Each workgroup: 8 waves = 4 (tile, khalf) pairs. Tile = 16 tokens (WMMA M).
// Wave w: tile = w>>1, khalf = w&1 covers K in [khalf*2048, khalf*2048+2048).
__global__ __launch_bounds__(256) void router_gemm_topk_kernel(
    const float* __restrict__ x, const float* __restrict__ W,
    const float* __restrict__ b, const float* __restrict__ noise,
    float* __restrict__ out, float* __restrict__ usage_ws)
{
    // W chunk staging, pair-interleaved: sW[h][(p*8+e)*2 + j] = W[base + 2p + j][e]
    __shared__ float sW[2][CHUNK * E_DIM];          // 2 x 16KB
    __shared__ float sLog[2 * 4 * 16 * E_DIM];      // [khalf][tile][token][expert], 4KB
    __shared__ float sUsage[E_DIM];

    const int tid   = threadIdx.x;
    const int wave  = tid >> 5;
    const int lane  = tid & 31;
    const int khalf = wave & 1;
    const int tile  = wave >> 1;
    const int tokBase = (blockIdx.x * 4 + tile) * 16;

    if (tid < E_DIM) sUsage[tid] = 0.0f;

    v8f acc = {};   // fp32 D accumulator: VGPR j, lanes 0-15 -> M=j, lanes 16-31 -> M=j+8

    const int row = tokBase + (lane & 15);
    const float* xrow = x + (size_t)row * H_DIM + (size_t)khalf * KHALF + ((lane >> 4) * 4);

    // rank of this thread among the 4 waves of its K-half (for cooperative W loads)
    const int rank = ((tid >> 6) << 5) | (tid & 31);   // 0..127
    const float* sB = &sW[khalf][(lane & 7) * 2];
    const int phalf = (lane >> 4) << 1;                // hi lanes read rows +4,+5

    for (int it = 0; it < KHALF / CHUNK; ++it) {
        const int kc = it * CHUNK;
        __syncthreads();
        // stage this half's W chunk into LDS pair layout (W is tiny & reused -> regular TH)
        const int kbase = khalf * KHALF + kc;
        for (int q = rank; q < (CHUNK / 2) * E_DIM; q += 128) {
            int p = q >> 3, e = q & 7;
            int k = kbase + 2 * p;
            float2 w2;
            w2.x = W[(size_t)k * E_DIM + e];
            w2.y = W[(size_t)(k + 1) * E_DIM + e];
            *(float2*)&sW[khalf][q * 2] = w2;
        }
        __syncthreads();

        #pragma unroll 8
        for (int c = 0; c < CHUNK; c += 8) {
            // x is a 256MB one-pass stream (> 192MB L2): non-temporal to avoid
            // evicting the hot W / noise working set.
            // lanes 0-15 get x[row][kc+c .. +3], lanes 16-31 get x[row][kc+c+4 .. +7]
            v4f a4 = __builtin_nontemporal_load((const v4f*)(xrow + kc + c));
            int p1 = (c >> 1) + phalf;
            v2f b1 = *(const v2f*)(sB + p1 * 16);        // rows (c, c+1) | (c+4, c+5)
            v2f b2 = *(const v2f*)(sB + (p1 + 1) * 16);  // rows (c+2,c+3) | (c+6,c+7)
            v2f aA = { a4.x, a4.y };
            v2f aB = { a4.z, a4.w };
            acc = __builtin_amdgcn_wmma_f32_16x16x4_f32(false, aA, false, b1,
                                                        (short)0, acc, false, false);
            acc = __builtin_amdgcn_wmma_f32_16x16x4_f32(false, aB, false, b2,
                                                        (short)0, acc, false, false);
        }
    }

    // spill partial logits: lane holds expert column e = lane&15 (e>=8 are dup columns, drop)
    {
        int e = lane & 15;
        if (e < E_DIM) {
            int thalf = (lane >> 4) * 8;
            float* dst = &sLog[(((khalf * 4 + tile) * 16) + thalf) * E_DIM + e];
            #pragma unroll
            for (int j = 0; j < 8; ++j) dst[j * E_DIM] = acc[j];
        }
    }
    __syncthreads();

    // phase 2: one thread per token (64 tokens per WG): combine split-K halves,
    // add bias + scaled noise, softmax over 8 experts, top-2 + renormalize.
    if (tid < 64) {
        const int ptile = tid >> 4;
        const int tl    = tid & 15;
        const int t     = blockIdx.x * 64 + tid;
        const float* l0 = &sLog[((0 * 4 + ptile) * 16 + tl) * E_DIM];
        const float* l1 = &sLog[((1 * 4 + ptile) * 16 + tl) * E_DIM];
        v4f n0 = *(const v4f*)(noise + (size_t)t * E_DIM);
        v4f n1 = *(const v4f*)(noise + (size_t)t * E_DIM + 4);
        float nz[8] = { n0.x, n0.y, n0.z, n0.w, n1.x, n1.y, n1.z, n1.w };

        float li[E_DIM];
        float m = -1e30f;
        #pragma unroll
        for (int e = 0; e < E_DIM; ++e) {
            li[e] = l0[e] + l1[e] + b[e] + NOISE_EPS * nz[e];
            m = fmaxf(m, li[e]);
        }
        float r[E_DIM];
        float s = 0.0f;
        #pragma unroll
        for (int e = 0; e < E_DIM; ++e) { r[e] = __expf(li[e] - m); s += r[e]; }
        float inv = 1.0f / s;
        #pragma unroll
        for (int e = 0; e < E_DIM; ++e) {
            r[e] *= inv;
            atomicAdd(&sUsage[e], r[e]);     // ds_add_f32
        }
        // top-2, lowest index wins ties (matches lax.top_k)
        int i1 = 0; float v1 = r[0];
        #pragma unroll
        for (int e = 1; e < E_DIM; ++e) if (r[e] > v1) { v1 = r[e]; i1 = e; }
        int i2 = -1; float v2 = -1e30f;
        #pragma unroll
        for (int e = 0; e < E_DIM; ++e) if (e != i1 && r[e] > v2) { v2 = r[e]; i2 = e; }
        float inv2 = 1.0f / (v1 + v2);
        out[(size_t)t * 2 + 0] = v1 * inv2;
        out[(size_t)t * 2 + 1] = v2 * inv2;
        out[(size_t)NTOK * 2 + (size_t)t * 2 + 0] = (float)i1;
        out[(size_t)NTOK * 2 + (size_t)t * 2 + 1] = (float)i2;
    }
    __syncthreads();
    if (tid < E_DIM) atomicAdd(&usage_ws[tid], sUsage[tid]);
}

__global__ void router_auxloss_kernel(const float* __restrict__ usage_ws,
                                      float* __restrict__ out)
{
    if (threadIdx.x == 0 && blockIdx.x == 0) {
        float s = 0.0f;
        #pragma unroll
        for (int e = 0; e < E_DIM; ++e) {
            float d = usage_ws[e] * (1.0f / (float)NTOK) - (1.0f / (float)E_DIM);
            s += d * d;
        }
        out[(size_t)NTOK * 4] = LB_COEFF * s;   // out[65536]
    }
}

extern "C" void kernel_launch(void* const* d_in, const int* in_sizes, int n_in,
                              void* d_out, int out_size, void* d_ws, size_t ws_size,
                              hipStream_t stream) {
    const float* x     = (const float*)d_in[0];   // [4,4096,4096]
    const float* W     = (const float*)d_in[1];   // [4096,8]
    const float* b     = (const float*)d_in[2];   // [8]
    const float* noise = (const float*)d_in[3];   // [16384,8]
    float* out = (float*)d_out;
    float* ws  = (float*)d_ws;

    hipMemsetAsync(ws, 0, E_DIM * sizeof(float), stream);
    router_gemm_topk_kernel<<<dim3(NTOK / 64), dim3(256), 0, stream>>>(x, W, b, noise, out, ws);
    router_auxloss_kernel<<<dim3(1), dim3(32), 0, stream>>>(ws, out);
}